// BlellochScan_42700564857293
// MI455X (gfx1250) — compile-verified
//
#include <hip/hip_runtime.h>

// Problem constants (from reference): X_in (B=8, L=4096, D=64, N=16) fp32.
// Task == inclusive cumsum along L for C = D*N = 1024 contiguous channels per (b,l).
#define B_   8
#define L_   4096
#define C_   1024
#define SEGS 32
#define SEGL (L_ / SEGS)   // 128

typedef __attribute__((ext_vector_type(2))) float v2f;
typedef __attribute__((ext_vector_type(8))) float v8f;

// ---------------- Pass 1: per-segment channel sums ----------------
// segsum[b][s][c] = sum_{l in segment s} X[b][l][c]
__global__ void seg_reduce_kernel(const float* __restrict__ x,
                                  float* __restrict__ segsum) {
    int t = blockIdx.x * blockDim.x + threadIdx.x;   // 0 .. B*SEGS*C-1
    int c = t % C_;
    int s = (t / C_) % SEGS;
    int b = t / (C_ * SEGS);
    const float* p = x + ((size_t)(b * L_ + s * SEGL)) * C_ + c;
    float acc = 0.0f;
#pragma unroll 8
    for (int l = 0; l < SEGL; ++l) acc += p[(size_t)l * C_];
    segsum[t] = acc;
}

// ---------------- Pass 2: exclusive scan of segment sums (in place) ----------------
__global__ void seg_scan_kernel(float* __restrict__ segsum) {
    int t = blockIdx.x * blockDim.x + threadIdx.x;   // 0 .. B*C-1
    int c = t % C_;
    int b = t / C_;
    float* p = segsum + (size_t)b * SEGS * C_ + c;
    float run = 0.0f;
#pragma unroll
    for (int s = 0; s < SEGS; ++s) {
        float v = p[(size_t)s * C_];
        p[(size_t)s * C_] = run;
        run += v;
    }
}

// ---------------- Pass 3: WMMA segmented scan ----------------
// One wave handles 16 channels x one 128-long segment. A 16-step scan block is
// D = T(16x16 lower-tri ones) * Xblk(16x16) + carry, realized as a chain of four
// V_WMMA_F32_16X16X4_F32 (K=4 each) accumulating through C.
//
// ISA operand layouts (wave32):
//   A 16x4 f32 : lane ln -> m = ln%16 ; VGPR v, group g=ln/16 -> k = 2g + v
//   B 4x16 f32 : lane ln -> n = ln%16 ; VGPR v, group g      -> k = 2g + v
//   C/D 16x16  : lane ln -> n = ln%16 ; VGPR v               -> m = v + 8g
// Carry out = row 15 of D = acc[7] in lanes 16..31 -> one shuffle.
__global__ void wmma_scan_kernel(const float* __restrict__ x,
                                 const float* __restrict__ segoff,
                                 float* __restrict__ out) {
    const int lane = threadIdx.x & 31;
    const int wid  = blockIdx.x * (blockDim.x >> 5) + (threadIdx.x >> 5);
    const int cb   = wid & 63;          // 64 channel blocks of 16
    const int seg  = (wid >> 6) & (SEGS - 1);
    const int b    = wid >> 11;         // 8 batches
    const int g    = lane >> 4;
    const int j    = lane & 15;         // column index (also m for A)
    const int c0   = cb * 16;

    // Lower-triangular ones operand, split into 4 K-chunks.
    v2f A[4];
#pragma unroll
    for (int s = 0; s < 4; ++s) {
#pragma unroll
        for (int v = 0; v < 2; ++v) {
            int k = 4 * s + 2 * g + v;          // global K index 0..15
            A[s][v] = (k <= j) ? 1.0f : 0.0f;   // inclusive scan
        }
    }

    // Running carry for this segment: exclusive prefix of earlier segments.
    float carry = segoff[((size_t)(b * SEGS + seg)) * C_ + c0 + j];

    const size_t base = ((size_t)(b * L_ + seg * SEGL)) * C_ + c0 + j;
    const float* xp = x + base;
    float*       op = out + base;

    for (int blk = 0; blk < SEGL / 16; ++blk) {
        const size_t blkOff = (size_t)blk * 16 * C_;

        v8f acc;
#pragma unroll
        for (int i = 0; i < 8; ++i) acc[i] = carry;   // C = carry broadcast per column

#pragma unroll
        for (int s = 0; s < 4; ++s) {
            v2f Bop;
#pragma unroll
            for (int v = 0; v < 2; ++v) {
                int k = 4 * s + 2 * g + v;
                Bop[v] = xp[blkOff + (size_t)k * C_];
            }
            // (neg_a, A, neg_b, B, c_mod, C, reuse_a, reuse_b)
            acc = __builtin_amdgcn_wmma_f32_16x16x4_f32(
                false, A[s], false, Bop, (short)0, acc, false, false);
        }

        // Store D: row m = v + 8g, col n = j.
#pragma unroll
        for (int v = 0; v < 8; ++v) {
            op[blkOff + (size_t)(v + 8 * g) * C_] = acc[v];
        }

        // New carry = D[15][j] = acc[7] held in lane 16 + j.
        carry = __shfl(acc[7], 16 + j, 32);
    }
}

extern "C" void kernel_launch(void* const* d_in, const int* in_sizes, int n_in,
                              void* d_out, int out_size, void* d_ws, size_t ws_size,
                              hipStream_t stream) {
    const float* x   = (const float*)d_in[0];
    float*       out = (float*)d_out;
    float*       seg = (float*)d_ws;           // B*SEGS*C floats = 1 MiB

    // Pass 1: segment reduction. B*SEGS*C = 262144 threads.
    {
        int total = B_ * SEGS * C_;
        seg_reduce_kernel<<<total / 256, 256, 0, stream>>>(x, seg);
    }
    // Pass 2: exclusive scan over segments per channel. B*C = 8192 threads.
    {
        int total = B_ * C_;
        seg_scan_kernel<<<total / 256, 256, 0, stream>>>(seg);
    }
    // Pass 3: WMMA scan. 8 b * 32 seg * 64 cb = 16384 waves, 8 waves/block.
    {
        int waves  = B_ * SEGS * 64;
        int blocks = waves / 8;
        wmma_scan_kernel<<<blocks, 256, 0, stream>>>(x, seg, out);
    }
}